// MQCNN_Conv2d_39470749450784
// MI455X (gfx1250) — compile-verified
//
#include <hip/hip_runtime.h>
#include <hip/hip_bf16.h>

// Sizes fixed by setup_inputs(): inputs (16,3,32,32) f32, weights (16,3,16) f32,
// output (16,16,31,31) f32.  Patches per channel N = 16*31*31 = 15376 = 961 tiles of 16.

typedef float v2f __attribute__((ext_vector_type(2)));
typedef float v8f __attribute__((ext_vector_type(8)));

struct Gate { signed char op, a, b, wi; };  // op 0 = RY(wire=a, theta=w[wi]); op 1 = CNOT(ctrl=a, tgt=b)

#define RY(wr, wi)  { 0, (wr), 0, (wi) }
#define CX(cc, tt)  { 1, (cc), (tt), 0 }

__device__ __constant__ Gate GATES[5][24] = {
  // 0: circ_ran (10 gates)
  { RY(0,0),RY(1,1),RY(2,2),RY(3,3), CX(0,1),CX(2,3), RY(0,4),RY(2,5), CX(1,2), RY(1,6),
    {0,0,0,0},{0,0,0,0},{0,0,0,0},{0,0,0,0},{0,0,0,0},{0,0,0,0},{0,0,0,0},
    {0,0,0,0},{0,0,0,0},{0,0,0,0},{0,0,0,0},{0,0,0,0},{0,0,0,0},{0,0,0,0} },
  // 1: circ_line (10 gates)
  { RY(0,0),RY(1,1),RY(2,2),RY(3,3), CX(0,1),CX(1,2),CX(2,3), RY(1,4),RY(2,5),RY(3,6),
    {0,0,0,0},{0,0,0,0},{0,0,0,0},{0,0,0,0},{0,0,0,0},{0,0,0,0},{0,0,0,0},
    {0,0,0,0},{0,0,0,0},{0,0,0,0},{0,0,0,0},{0,0,0,0},{0,0,0,0},{0,0,0,0} },
  // 2: circ_ring (12 gates)
  { RY(0,0),RY(1,1),RY(2,2),RY(3,3), CX(0,1),CX(1,2),CX(2,3),CX(3,0),
    RY(0,4),RY(1,5),RY(2,6),RY(3,7),
    {0,0,0,0},{0,0,0,0},{0,0,0,0},{0,0,0,0},{0,0,0,0},{0,0,0,0},
    {0,0,0,0},{0,0,0,0},{0,0,0,0},{0,0,0,0},{0,0,0,0},{0,0,0,0} },
  // 3: circ_doublering (24 gates)
  { RY(0,0),RY(1,1),RY(2,2),RY(3,3), CX(0,1),CX(1,2),CX(2,3),CX(3,0),
    RY(0,4),RY(1,5),RY(2,6),RY(3,7), CX(3,0),CX(2,3),CX(1,2),CX(0,1),
    RY(0,8),RY(1,9),RY(2,10),RY(3,11), CX(0,1),CX(1,2),CX(2,3),CX(3,0) },
  // layer-2 tail of doublering handled by extension table below (needs 32 gates total);
  // see GATES_DR_TAIL for the remainder.
  // 4: circ_blockring (24 gates)
  { RY(0,0),RY(1,1),RY(2,2),RY(3,3), CX(0,1),CX(2,3),CX(1,2),CX(3,0),
    RY(0,4),RY(1,5),RY(2,6),RY(3,7), CX(0,1),CX(2,3),CX(1,2),CX(3,0),
    RY(0,8),RY(1,9),RY(2,10),RY(3,11), CX(0,1),CX(2,3),CX(1,2),CX(3,0) },
};

// Remainder of circ_doublering (it has 2 layers x 16 gates = 32 gates total;
// first 24 are in GATES[3], the last 8 are here).
__device__ __constant__ Gate GATES_DR_TAIL[8] = {
  RY(0,12),RY(1,13),RY(2,14),RY(3,15), CX(3,0),CX(2,3),CX(1,2),CX(0,1)
};

__device__ __constant__ int GCNT[5] = { 10, 10, 12, 24 /* + 8 tail */, 24 };

// ---------------------------------------------------------------------------
// Kernel 1: build M_{i,c} = U^T diag(sign) U  for the 48 (i,c) pairs.
// One block of 256 threads per pair; U[16][16] simulated in LDS by applying
// the gate list to the columns of the identity.  Wire w <-> bit (3-w).
// ---------------------------------------------------------------------------
__global__ void __launch_bounds__(256) build_M_kernel(const float* __restrict__ wts,
                                                      float* __restrict__ Mout) {
  const int pair = blockIdx.x;          // 0..47  (= i*3 + c)
  const int i    = pair / 3;
  const int t    = threadIdx.x;         // 0..255
  const int m    = t >> 4;              // state index (row)
  const int j    = t & 15;              // column
  const int circ = i % 5;

  __shared__ float U[16][17];
  U[m][j] = (m == j) ? 1.0f : 0.0f;
  __syncthreads();

  const float* wv = wts + pair * 16;

  const int ng = GCNT[circ];
  for (int g = 0; g < ng + ((circ == 3) ? 8 : 0); ++g) {
    Gate gd = (g < 24) ? GATES[circ][g] : GATES_DR_TAIL[g - 24];
    if (gd.op == 0) {
      // RY(theta) on wire gd.a:  out0 = c*in0 - s*in1 ; out1 = s*in0 + c*in1
      const int  mask = 1 << (3 - gd.a);
      const float th  = 0.5f * wv[(int)gd.wi];
      float sh, ch;  sincosf(th, &sh, &ch);
      const float um = U[m][j];
      const float up = U[m ^ mask][j];
      __syncthreads();
      U[m][j] = ((m & mask) == 0) ? (ch * um - sh * up)
                                  : (sh * up + ch * um);
      __syncthreads();
    } else {
      // CNOT(ctrl=gd.a, tgt=gd.b): out[m] = in[m ^ (ctrl_set ? tgt_mask : 0)]
      const int cb = 3 - gd.a, tb = 3 - gd.b;
      const int src = m ^ (((m >> cb) & 1) << tb);
      const float v = U[src][j];
      __syncthreads();
      U[m][j] = v;
      __syncthreads();
    }
  }

  // M[j][k] = sum_m sign(m) * U[m][j] * U[m][k],  sign = +1 iff m < 8 (wire-0 bit)
  float acc = 0.0f;
#pragma unroll
  for (int mm = 0; mm < 16; ++mm) {
    const float sgn = (mm < 8) ? 1.0f : -1.0f;
    acc += sgn * U[mm][m] * U[mm][j];
  }
  Mout[(pair * 16 + m) * 16 + j] = acc;   // row-major 16x16 per pair
}

// ---------------------------------------------------------------------------
// Kernel 2: per (patch-tile of 16, output channel i):
//   for c in 0..2:  S = embedding tile (16x16),  P = S * M_{i,c} via 4 chained
//   v_wmma_f32_16x16x4_f32,  part[row] += rowsum(P .* S).
// One wave (32 threads) per block.  961 exact tiles, grid (961, 16).
//
// f32 WMMA fragment layouts (CDNA5 ISA):
//   A (16x4): lane L<16 -> row L, v0=K0, v1=K1 ; lane 16+L -> row L, v0=K2, v1=K3
//   B (4x16): v0: lanes0-15 = row K0 (N=lane), lanes16-31 = row K2 ; v1: K1 / K3
//   C/D (16x16): vgpr r: lanes0-15 -> (M=r, N=lane), lanes16-31 -> (M=r+8, N=lane-16)
// ---------------------------------------------------------------------------
__global__ void __launch_bounds__(32) quanv_wmma_kernel(const float* __restrict__ in,
                                                        const float* __restrict__ M,
                                                        float* __restrict__ out) {
  const int tile = blockIdx.x;          // 0..960
  const int oi   = blockIdx.y;          // 0..15
  const int lane = threadIdx.x;         // 0..31
  const int half = lane >> 4;
  const int col  = lane & 15;
  const int row  = lane & 15;

  __shared__ float S[16][17];

  float part[8];
#pragma unroll
  for (int r = 0; r < 8; ++r) part[r] = 0.0f;

  for (int c = 0; c < 3; ++c) {
    __syncthreads();
    if (lane < 16) {
      // Build one embedding row: patch p = tile*16 + lane
      const int p   = tile * 16 + lane;       // always < 15376 (961*16 exact)
      const int b   = p / 961;
      const int rem = p % 961;
      const int h   = rem / 31;
      const int w   = rem % 31;
      const float* base = in + ((b * 3 + c) * 32 + h) * 32 + w;
      const float x0 = base[0],  x1 = base[1];
      const float x2 = base[32], x3 = base[33];
      float sh0, ch0, sh1, ch1, sh2, ch2, sh3, ch3;
      sincosf(0.5f * x0, &sh0, &ch0);
      sincosf(0.5f * x1, &sh1, &ch1);
      sincosf(0.5f * x2, &sh2, &ch2);
      sincosf(0.5f * x3, &sh3, &ch3);
#pragma unroll
      for (int mm = 0; mm < 16; ++mm) {
        // wire0 -> bit3, wire1 -> bit2, wire2 -> bit1, wire3 -> bit0
        const float v = ((mm & 8) ? sh0 : ch0) * ((mm & 4) ? sh1 : ch1)
                      * ((mm & 2) ? sh2 : ch2) * ((mm & 1) ? sh3 : ch3);
        S[lane][mm] = v;
      }
    }
    __syncthreads();

    const float* Mp = M + (oi * 3 + c) * 256;

    v8f acc = {};
#pragma unroll
    for (int s = 0; s < 4; ++s) {
      const int k0 = 4 * s;
      v2f a, bf;
      a[0]  = S[row][k0 + 2 * half + 0];
      a[1]  = S[row][k0 + 2 * half + 1];
      bf[0] = Mp[(k0 + 2 * half + 0) * 16 + col];
      bf[1] = Mp[(k0 + 2 * half + 1) * 16 + col];
      acc = __builtin_amdgcn_wmma_f32_16x16x4_f32(
                false, a, false, bf, (short)0, acc, false, false);
    }

    // part[r] += P[row_r][col] * S[row_r][col]   (row_r = r + 8*half)
#pragma unroll
    for (int r = 0; r < 8; ++r) {
      part[r] += acc[r] * S[r + 8 * half][col];
    }
  }

  // Horizontal reduction over the 16 columns held across each 16-lane half.
#pragma unroll
  for (int r = 0; r < 8; ++r) {
    float v = part[r];
    v += __shfl_xor(v, 1);
    v += __shfl_xor(v, 2);
    v += __shfl_xor(v, 4);
    v += __shfl_xor(v, 8);
    part[r] = v;   // lanes 0-15: rowsum(r); lanes 16-31: rowsum(r+8)
  }

  float outv = 0.0f;
#pragma unroll
  for (int r = 0; r < 8; ++r) {
    const float vl = __shfl(part[r], 0);    // rowsum(r)
    const float vh = __shfl(part[r], 16);   // rowsum(r+8)
    if (lane == r)     outv = vl;
    if (lane == r + 8) outv = vh;
  }

  if (lane < 16) {
    const int p   = tile * 16 + lane;
    const int b   = p / 961;
    const int rem = p % 961;
    // out[b][oi][h][w] -> ((b*16 + oi)*31 + h)*31 + w = (b*16+oi)*961 + rem
    out[(b * 16 + oi) * 961 + rem] = outv;
  }
}

extern "C" void kernel_launch(void* const* d_in, const int* in_sizes, int n_in,
                              void* d_out, int out_size, void* d_ws, size_t ws_size,
                              hipStream_t stream) {
  const float* inputs  = (const float*)d_in[0];   // (16,3,32,32) f32
  const float* weights = (const float*)d_in[1];   // (16,3,16)  f32
  float*       out     = (float*)d_out;           // (16,16,31,31) f32
  float*       Mbuf    = (float*)d_ws;            // 48 * 16 * 16 f32 = 48 KB

  build_M_kernel<<<48, 256, 0, stream>>>(weights, Mbuf);
  quanv_wmma_kernel<<<dim3(961, 16), 32, 0, stream>>>(inputs, Mbuf, out);
}